// input_layer_78323023610512
// MI455X (gfx1250) — compile-verified
//
#include <hip/hip_runtime.h>
#include <hip/hip_bf16.h>
#include <math.h>

// ---------------------------------------------------------------------------
// SPDNet forward for MI455X (gfx1250, wave32).
//   stage 1: M1 = W1^T X W1   (57->20)  via V_WMMA_F32_16X16X4_F32, 4 waves/WG
//   ReEig(20x20)  batched cyclic Jacobi, one wave per matrix
//   stage 2: M2 = W2^T M1' W2 (20->10)  via WMMA f32, 2 waves/WG
//   ReEig(10x10)  batched Jacobi
//   stage 3+ReEig(2x2): closed form, one matrix per lane
//
// Roofline: X stream = 32768*57*57*4 B ~ 429 MB -> ~18 us floor @ 23.3 TB/s;
// bilinear FLOPs (~6 GFLOP fp32) are negligible vs the WMMA ceiling, so the
// transform kernels are built for bandwidth (NT loads, multi-wave WGs), and
// fp32 WMMA is used to preserve the precision the eigensolve needs.
// ---------------------------------------------------------------------------

typedef __attribute__((ext_vector_type(2))) float v2f;
typedef __attribute__((ext_vector_type(8))) float v8f;

#define SPD_EPS 1e-4f

// ---------------------------------------------------------------------------
// Bilinear SPD transform: M = W^T X W, WAVES wave32s per batch matrix.
// X: [NIN,NIN], W: [NIN,NOUT], M: [NOUT,NOUT].  Padded in LDS to mult-of-16.
// ---------------------------------------------------------------------------
template <int NIN, int NOUT, int WAVES>
__global__ __launch_bounds__(32 * WAVES) void spd_bilinear(
    const float* __restrict__ Xg, const float* __restrict__ Wg,
    float* __restrict__ Mg) {
  constexpr int KP = (NIN + 15) & ~15;   // padded inner dim
  constexpr int NP = (NOUT + 15) & ~15;  // padded outer dim
  constexpr int KPT = KP / 16;           // tile counts
  constexpr int NPT = NP / 16;
  constexpr int NT = 32 * WAVES;

  __shared__ float Xs[KP][KP];
  __shared__ float Ws[KP][NP];
  __shared__ float Ts[KP][NP];

  const int b = blockIdx.x;
  const int tid = threadIdx.x;
  const int lane = tid & 31;
  // Provably wave-uniform wave id: keeps the tile loops scalar-branched so
  // EXEC is structurally all-ones at every WMMA (ISA requirement).
  const int wave = __builtin_amdgcn_readfirstlane(tid >> 5);
  const float* X = Xg + (size_t)b * NIN * NIN;

  // Cooperative LDS fill, zero-padded.  X is streamed once (>> L2) -> NT hint.
  for (int idx = tid; idx < KP * KP; idx += NT) {
    int r = idx / KP, c = idx % KP;
    Xs[r][c] = (r < NIN && c < NIN)
                   ? __builtin_nontemporal_load(&X[r * NIN + c])
                   : 0.0f;
  }
  for (int idx = tid; idx < KP * NP; idx += NT) {
    int r = idx / NP, c = idx % NP;
    Ws[r][c] = (r < NIN && c < NOUT) ? Wg[r * NOUT + c] : 0.0f;
  }
  __syncthreads();

  const int mrow = lane & 15;     // M (or N) index within a 16-tile
  const int khalf = (lane >> 4);  // 0: K pair {k0,k0+1}; 1: {k0+2,k0+3}

  // ---------------- Stage A: T = X * W (tiles split across waves) ----------
  for (int t = wave; t < KPT * NPT; t += WAVES) {  // scalar (uniform) trip
    const int ti = t / NPT, tj = t % NPT;
    v8f acc = {};
    for (int k0 = 0; k0 < KP; k0 += 4) {
      const int ka = k0 + 2 * khalf;
      v2f a, bb;
      a.x = Xs[ti * 16 + mrow][ka];
      a.y = Xs[ti * 16 + mrow][ka + 1];
      bb.x = Ws[ka][tj * 16 + mrow];
      bb.y = Ws[ka + 1][tj * 16 + mrow];
      acc = __builtin_amdgcn_wmma_f32_16x16x4_f32(
          false, a, false, bb, (short)0, acc, false, false);
    }
#pragma unroll
    for (int r = 0; r < 8; ++r)
      Ts[ti * 16 + r + 8 * khalf][tj * 16 + mrow] = acc[r];
  }
  __syncthreads();

  // ---------------- Stage B: M = W^T * T ----------------
  float* M = Mg + (size_t)b * NOUT * NOUT;
  for (int t = wave; t < NPT * NPT; t += WAVES) {
    const int ti = t / NPT, tj = t % NPT;
    v8f acc = {};
    for (int k0 = 0; k0 < KP; k0 += 4) {
      const int ka = k0 + 2 * khalf;
      v2f a, bb;
      a.x = Ws[ka][ti * 16 + mrow];  // W^T[m][k] = W[k][m]
      a.y = Ws[ka + 1][ti * 16 + mrow];
      bb.x = Ts[ka][tj * 16 + mrow];
      bb.y = Ts[ka + 1][tj * 16 + mrow];
      acc = __builtin_amdgcn_wmma_f32_16x16x4_f32(
          false, a, false, bb, (short)0, acc, false, false);
    }
#pragma unroll
    for (int r = 0; r < 8; ++r) {
      const int i = ti * 16 + r + 8 * khalf;
      const int j = tj * 16 + mrow;
      if (i < NOUT && j < NOUT) M[i * NOUT + j] = acc[r];
    }
  }
}

// Round-robin tournament position: fixes element 0, rotates the rest.
__device__ __forceinline__ int tpos(int k, int rnd, int n) {
  return (k == 0) ? 0 : 1 + ((k - 1 + rnd) % (n - 1));
}

// ---------------------------------------------------------------------------
// ReEig via cyclic Jacobi, one wave32 per NxN symmetric matrix, in place.
// buf[b] <- U * diag(max(eig, eps)) * U^T
// ---------------------------------------------------------------------------
template <int N, int SWEEPS>
__global__ __launch_bounds__(32) void jacobi_reeig(float* __restrict__ buf) {
  constexpr int H = N / 2;
  __shared__ float A[N][N + 1];
  __shared__ float U[N][N + 1];
  __shared__ float rc[H];
  __shared__ float rs[H];

  const int b = blockIdx.x;
  const int lane = threadIdx.x;
  float* M = buf + (size_t)b * N * N;

  for (int idx = lane; idx < N * N; idx += 32) {
    A[idx / N][idx % N] = M[idx];
    U[idx / N][idx % N] = (idx / N == idx % N) ? 1.0f : 0.0f;
  }
  __syncthreads();

  for (int sw = 0; sw < SWEEPS; ++sw) {
    for (int rnd = 0; rnd < N - 1; ++rnd) {
      // lanes 0..H-1: compute rotation (c,s) for their disjoint pair
      if (lane < H) {
        const int p = tpos(lane, rnd, N);
        const int q = tpos(N - 1 - lane, rnd, N);
        const float app = A[p][p], aqq = A[q][q], apq = A[p][q];
        float c = 1.0f, s = 0.0f;
        if (fabsf(apq) > 1e-30f) {
          const float tau = (aqq - app) / (2.0f * apq);
          const float t = copysignf(1.0f, tau) /
                          (fabsf(tau) + sqrtf(1.0f + tau * tau));
          c = 1.0f / sqrtf(1.0f + t * t);
          s = t * c;
        }
        rc[lane] = c;
        rs[lane] = s;
      }
      __syncthreads();
      // column rotations (A <- A*J) + eigenvector accumulation (U <- U*J);
      // lane j owns row j, pairs are disjoint so the loop is race-free.
      if (lane < N) {
        for (int i = 0; i < H; ++i) {
          const int p = tpos(i, rnd, N);
          const int q = tpos(N - 1 - i, rnd, N);
          const float c = rc[i], s = rs[i];
          float x = A[lane][p], y = A[lane][q];
          A[lane][p] = c * x - s * y;
          A[lane][q] = s * x + c * y;
          x = U[lane][p];
          y = U[lane][q];
          U[lane][p] = c * x - s * y;
          U[lane][q] = s * x + c * y;
        }
      }
      __syncthreads();
      // row rotations (A <- J^T*A); lane j owns column j.
      if (lane < N) {
        for (int i = 0; i < H; ++i) {
          const int p = tpos(i, rnd, N);
          const int q = tpos(N - 1 - i, rnd, N);
          const float c = rc[i], s = rs[i];
          const float x = A[p][lane], y = A[q][lane];
          A[p][lane] = c * x - s * y;
          A[q][lane] = s * x + c * y;
        }
      }
      __syncthreads();
    }
  }

  // Reconstruct with floored eigenvalues: R = U diag(max(diag(A),eps)) U^T
  if (lane < N) {
    for (int i = 0; i < N; ++i) {
      float acc = 0.0f;
      for (int k = 0; k < N; ++k)
        acc += U[i][k] * fmaxf(A[k][k], SPD_EPS) * U[lane][k];
      M[i * N + lane] = acc;
    }
  }
}

// ---------------------------------------------------------------------------
// Stage 3: Y = W3^T X3 W3 (10->2) + closed-form 2x2 ReEig.  One matrix/lane.
// ---------------------------------------------------------------------------
__global__ __launch_bounds__(256) void stage3_reeig2(const float* __restrict__ X3,
                                                     const float* __restrict__ W3,
                                                     float* __restrict__ out,
                                                     int B) {
  const int b = blockIdx.x * blockDim.x + threadIdx.x;
  if (b >= B) return;
  const float* X = X3 + (size_t)b * 100;

  float t0[10], t1[10];
#pragma unroll
  for (int k = 0; k < 10; ++k) {
    float s0 = 0.0f, s1 = 0.0f;
#pragma unroll
    for (int m = 0; m < 10; ++m) {
      const float x = X[k * 10 + m];
      s0 = fmaf(x, W3[m * 2 + 0], s0);
      s1 = fmaf(x, W3[m * 2 + 1], s1);
    }
    t0[k] = s0;
    t1[k] = s1;
  }
  float y00 = 0.0f, y01 = 0.0f, y10 = 0.0f, y11 = 0.0f;
#pragma unroll
  for (int k = 0; k < 10; ++k) {
    const float w0 = W3[k * 2 + 0], w1 = W3[k * 2 + 1];
    y00 = fmaf(w0, t0[k], y00);
    y01 = fmaf(w0, t1[k], y01);
    y10 = fmaf(w1, t0[k], y10);
    y11 = fmaf(w1, t1[k], y11);
  }

  // 2x2 symmetric ReEig
  const float a = y00, c = y11, bq = 0.5f * (y01 + y10);
  const float mean = 0.5f * (a + c), diff = 0.5f * (a - c);
  const float rad = sqrtf(diff * diff + bq * bq);
  const float f1 = fmaxf(mean + rad, SPD_EPS);
  const float f2 = fmaxf(mean - rad, SPD_EPS);

  float r00, r01, r11;
  if (fabsf(bq) > 1e-30f) {
    float vx = bq, vy = (mean + rad) - a;  // eigvec of larger eig
    float nrm = vx * vx + vy * vy;
    if (nrm < 1e-30f) {
      vx = 1.0f;
      vy = 0.0f;
      nrm = 1.0f;
    }
    const float inv = 1.0f / nrm;
    r00 = (f1 * vx * vx + f2 * vy * vy) * inv;
    r01 = (f1 - f2) * vx * vy * inv;
    r11 = (f1 * vy * vy + f2 * vx * vx) * inv;
  } else {
    r00 = fmaxf(a, SPD_EPS);
    r01 = 0.0f;
    r11 = fmaxf(c, SPD_EPS);
  }
  float* o = out + (size_t)b * 4;
  o[0] = r00;
  o[1] = r01;
  o[2] = r01;
  o[3] = r11;
}

// ---------------------------------------------------------------------------
extern "C" void kernel_launch(void* const* d_in, const int* in_sizes, int n_in,
                              void* d_out, int out_size, void* d_ws, size_t ws_size,
                              hipStream_t stream) {
  const float* x  = (const float*)d_in[0];  // [B,57,57]
  const float* W1 = (const float*)d_in[1];  // [57,20]
  const float* W2 = (const float*)d_in[2];  // [20,10]
  const float* W3 = (const float*)d_in[3];  // [10,2]
  float* out = (float*)d_out;               // [B,2,2]

  const int B = in_sizes[0] / (57 * 57);

  float* bufA = (float*)d_ws;            // B*400: M1 / rectified M1 (in place)
  float* bufB = bufA + (size_t)B * 400;  // B*100: M2 / rectified M2 (in place)

  spd_bilinear<57, 20, 4><<<B, 128, 0, stream>>>(x, W1, bufA);
  jacobi_reeig<20, 8><<<B, 32, 0, stream>>>(bufA);
  spd_bilinear<20, 10, 2><<<B, 64, 0, stream>>>(bufA, W2, bufB);
  jacobi_reeig<10, 7><<<B, 32, 0, stream>>>(bufB);
  stage3_reeig2<<<(B + 255) / 256, 256, 0, stream>>>(bufB, W3, out, B);
}